// QuantumConvLayer_85134841741326
// MI455X (gfx1250) — compile-verified
//
#include <hip/hip_runtime.h>
#include <math.h>

// ext-vector types for WMMA operands (f32 16x16x4: A=16x4 -> 2 VGPR, B=4x16 -> 2 VGPR, C/D -> 8 VGPR)
typedef __attribute__((ext_vector_type(2))) float v2f;
typedef __attribute__((ext_vector_type(8))) float v8f;

// ---------- monotonic float<->uint key mapping for min/max atomics ----------
__device__ __forceinline__ unsigned fkey(float f) {
  unsigned u = __float_as_uint(f);
  return (u & 0x80000000u) ? ~u : (u | 0x80000000u);
}
__device__ __forceinline__ float funkey(unsigned k) {
  unsigned u = (k & 0x80000000u) ? (k & 0x7FFFFFFFu) : ~k;
  return __uint_as_float(u);
}

// ---------- kernel 0: init atomic targets ----------
__global__ void qc_init(unsigned* keys) {
  keys[0] = 0xFFFFFFFFu;  // min key (identity = +inf key)
  keys[1] = 0u;           // max key (identity = -inf key)
}

// ---------- kernel 1: global min/max over x[:,0] (2,097,152 floats) ----------
__global__ __launch_bounds__(256) void qc_minmax(const float* __restrict__ x,
                                                 unsigned* __restrict__ keys,
                                                 int nvec4) {
  int idx = blockIdx.x * blockDim.x + threadIdx.x;
  int stride = gridDim.x * blockDim.x;
  float lmin = 3.402823466e38f, lmax = -3.402823466e38f;
  for (int e = idx; e < nvec4; e += stride) {
    int bb = e >> 14;  // 16384 float4 per (batch, channel0) plane
    int r4 = e & 16383;
    const float4* p = reinterpret_cast<const float4*>(x) + ((size_t)bb * 65536 + r4);
    float4 v = *p;
    lmin = fminf(lmin, fminf(fminf(v.x, v.y), fminf(v.z, v.w)));
    lmax = fmaxf(lmax, fmaxf(fmaxf(v.x, v.y), fmaxf(v.z, v.w)));
  }
#pragma unroll
  for (int off = 16; off > 0; off >>= 1) {
    lmin = fminf(lmin, __shfl_xor(lmin, off, 32));
    lmax = fmaxf(lmax, __shfl_xor(lmax, off, 32));
  }
  if ((threadIdx.x & 31) == 0) {
    atomicMin(&keys[0], fkey(lmin));
    atomicMax(&keys[1], fkey(lmax));
  }
}

// ---------- kernel 2: build A_q = Re(W^dag Z_q W) from weights ----------
// 256 threads: thread (k = tid>>4, t = tid&15) owns amplitude k of basis column |t>.
// Each RX/RY/CNOT gate is one LDS exchange round; RZ is local.
__global__ __launch_bounds__(256) void qc_build(const float* __restrict__ w, float* __restrict__ M) {
  __shared__ float Sr[16][16];  // Sr[amplitude][column]
  __shared__ float Si[16][16];
  int tid = threadIdx.x;
  int t = tid & 15;   // basis column
  int k = tid >> 4;   // amplitude index
  float ar = (k == t) ? 1.f : 0.f;
  float ai = 0.f;

  for (int l = 0; l < 2; l++) {
    for (int q = 0; q < 4; q++) {
      int bp = 1 << (3 - q);
      int kp = k ^ bp;
      float lo = (k & bp) ? 1.f : -1.f;  // -1 for upper (bit=0), +1 for lower (bit=1)
      float a0 = w[l * 12 + q * 3 + 0] * 0.5f;
      float a1 = w[l * 12 + q * 3 + 1] * 0.5f;
      float a2 = w[l * 12 + q * 3 + 2] * 0.5f;
      // RX: new = c*cur - i*s*partner (same form both halves)
      {
        float c = cosf(a0), s = sinf(a0);
        __syncthreads();
        Sr[k][t] = ar; Si[k][t] = ai;
        __syncthreads();
        float br = Sr[kp][t], bi = Si[kp][t];
        float nr = c * ar + s * bi;
        float ni = c * ai - s * br;
        ar = nr; ai = ni;
      }
      // RY: new = c*cur + sign*s*partner, sign = (bit ? +1 : -1)
      {
        float c = cosf(a1), s = sinf(a1);
        __syncthreads();
        Sr[k][t] = ar; Si[k][t] = ai;
        __syncthreads();
        float br = Sr[kp][t], bi = Si[kp][t];
        float ss = lo * s;
        float nr = c * ar + ss * br;
        float ni = c * ai + ss * bi;
        ar = nr; ai = ni;
      }
      // RZ: multiply by (c -/+ i*s): bit=0 -> (c - i s), bit=1 -> (c + i s). Local.
      {
        float c = cosf(a2), s = sinf(a2);
        float ss = lo * s;  // bit=0: -s, bit=1: +s
        float nr = c * ar - ss * ai;
        float ni = c * ai + ss * ar;
        ar = nr; ai = ni;
      }
    }
    // CNOT ring (0,1),(1,2),(2,3),(3,0): if control bit set, take amplitude with target bit flipped
    for (int q = 0; q < 4; q++) {
      int bc = 1 << (3 - q);
      int bt = 1 << (3 - ((q + 1) & 3));
      __syncthreads();
      Sr[k][t] = ar; Si[k][t] = ai;
      __syncthreads();
      int src = (k & bc) ? (k ^ bt) : k;
      ar = Sr[src][t]; ai = Si[src][t];
    }
  }
  // Publish final W columns, then A_q[i][t] = sum_k z_qk*(Wr[k][i]Wr[k][t]+Wi[k][i]Wi[k][t])
  __syncthreads();
  Sr[k][t] = ar; Si[k][t] = ai;
  __syncthreads();
  float acc0 = 0.f, acc1 = 0.f, acc2 = 0.f, acc3 = 0.f;
#pragma unroll
  for (int kk = 0; kk < 16; kk++) {
    float pr = Sr[kk][k] * Sr[kk][t] + Si[kk][k] * Si[kk][t];
    acc0 += ((kk >> 3) & 1) ? -pr : pr;
    acc1 += ((kk >> 2) & 1) ? -pr : pr;
    acc2 += ((kk >> 1) & 1) ? -pr : pr;
    acc3 += (kk & 1) ? -pr : pr;
  }
  M[0 * 256 + k * 16 + t] = acc0;
  M[1 * 256 + k * 16 + t] = acc1;
  M[2 * 256 + k * 16 + t] = acc2;
  M[3 * 256 + k * 16 + t] = acc3;
}

// ---------- kernel 3: main pass — encode + 4 quadratic forms via f32 WMMA ----------
// Each wave handles a tile of 16 samples. Per q: D = A_q (16x16) * Psi (16x16samples),
// built as 4 chained V_WMMA_F32_16X16X4_F32; then ev_q[n] = sum_m Psi[m,n]*D[m,n].
// NOTE: no private arrays anywhere — psi values are factored as ab[.]*cd[.] with
// named scalars so lane-half selection is a single v_cndmask on values (clang turns
// ternaries over array elements into dynamic-index compare/select trees otherwise).
__global__ __launch_bounds__(256) void qc_main(const float* __restrict__ x,
                                               const float* __restrict__ Mg,
                                               const unsigned* __restrict__ keys,
                                               float* __restrict__ out) {
  __shared__ float Ms[1024];  // A_q matrices: Ms[q*256 + m*16 + k]
  int tid = threadIdx.x;
#pragma unroll
  for (int i = tid; i < 1024; i += 256) Ms[i] = Mg[i];
  __syncthreads();

  float pmin = funkey(keys[0]);
  float pmax = funkey(keys[1]);
  float scale = 1.57079632679489662f / (pmax - pmin + 1e-8f);  // (pi/2) * norm

  int lane = tid & 31;
  int wave = tid >> 5;
  int tile = blockIdx.x * 8 + wave;  // 16 samples per tile
  int sidx = lane & 15;              // this lane's sample column within tile
  bool h = (lane & 16) != 0;         // lane half selects K/M sub-blocks
  int n = tile * 16 + sidx;          // global sample id, < 524288 exactly
  int b = n >> 14;                   // batch
  int p = n & 16383;
  int oi = p >> 7;    // output row
  int oj = p & 127;   // output col

  // patch = 2x2 block of x[b,0]: (2i,2j),(2i,2j+1),(2i+1,2j),(2i+1,2j+1)
  const float* row0 = x + (size_t)b * 262144 + (size_t)(2 * oi) * 256 + (size_t)(2 * oj);
  float2 r0 = *reinterpret_cast<const float2*>(row0);
  float2 r1 = *reinterpret_cast<const float2*>(row0 + 256);

  // Angles in [0, pi/2]: native v_sin/v_cos are accurate here.
  float t0 = (r0.x - pmin) * scale, t1 = (r0.y - pmin) * scale;
  float t2 = (r1.x - pmin) * scale, t3 = (r1.y - pmin) * scale;
  float c0, s0, c1, s1, c2, s2, c3, s3;
  __sincosf(t0, &s0, &c0);
  __sincosf(t1, &s1, &c1);
  __sincosf(t2, &s2, &c2);
  __sincosf(t3, &s3, &c3);

  // psi_k = ab[k>>2] * cd[k&3] where ab/cd are the qubit-pair cos/sin products.
  float ab0 = c0 * c1, ab1 = c0 * s1, ab2 = s0 * c1, ab3 = s0 * s1;
  float cd0 = c2 * c3, cd1 = c2 * s3, cd2 = s2 * c3, cd3 = s2 * s3;

  // Lane-half factors (4 value selects total):
  float cdx = h ? cd2 : cd0;  // cd[2*hf]
  float cdy = h ? cd3 : cd1;  // cd[2*hf+1]
  float abx = h ? ab2 : ab0;  // ab[2*hf]
  float aby = h ? ab3 : ab1;  // ab[2*hf+1]

  // B operand chunks (4x16 f32): Bop[kc] = { psi[kc*4+2hf], psi[kc*4+2hf+1] } = ab[kc]*{cdx,cdy}
  v2f B0 = {ab0 * cdx, ab0 * cdy};
  v2f B1 = {ab1 * cdx, ab1 * cdy};
  v2f B2 = {ab2 * cdx, ab2 * cdy};
  v2f B3 = {ab3 * cdx, ab3 * cdy};

  // Psi rows in C/D layout: prow[v] = psi[v+8hf] = ab[(v>>2)+2hf] * cd[v&3]
  float pr0 = abx * cd0, pr1 = abx * cd1, pr2 = abx * cd2, pr3 = abx * cd3;
  float pr4 = aby * cd0, pr5 = aby * cd1, pr6 = aby * cd2, pr7 = aby * cd3;

#pragma unroll
  for (int q = 0; q < 4; q++) {
    int abase = q * 256 + sidx * 16 + (h ? 2 : 0);
    v2f A0, A1, A2, A3;  // A operand (16x4): lane m=sidx, VGPR0=A[m,k0+2hf], VGPR1=A[m,k0+2hf+1]
    A0.x = Ms[abase + 0];  A0.y = Ms[abase + 1];
    A1.x = Ms[abase + 4];  A1.y = Ms[abase + 5];
    A2.x = Ms[abase + 8];  A2.y = Ms[abase + 9];
    A3.x = Ms[abase + 12]; A3.y = Ms[abase + 13];

    v8f acc = {0.f, 0.f, 0.f, 0.f, 0.f, 0.f, 0.f, 0.f};
    acc = __builtin_amdgcn_wmma_f32_16x16x4_f32(false, A0, false, B0, (short)0, acc, false, false);
    acc = __builtin_amdgcn_wmma_f32_16x16x4_f32(false, A1, false, B1, (short)0, acc, false, false);
    acc = __builtin_amdgcn_wmma_f32_16x16x4_f32(false, A2, false, B2, (short)0, acc, false, false);
    acc = __builtin_amdgcn_wmma_f32_16x16x4_f32(false, A3, false, B3, (short)0, acc, false, false);

    // ev_q[n] = sum_m psi[m]*D[m,n]
    float partial = pr0 * acc[0] + pr1 * acc[1] + pr2 * acc[2] + pr3 * acc[3] +
                    pr4 * acc[4] + pr5 * acc[5] + pr6 * acc[6] + pr7 * acc[7];
    partial += __shfl_xor(partial, 16, 32);  // combine the two lane halves (rows 0-7 + 8-15)
    if (!h) {
      out[(((size_t)b * 4 + q) * 128 + oi) * 128 + oj] = partial;
    }
  }
}

extern "C" void kernel_launch(void* const* d_in, const int* in_sizes, int n_in,
                              void* d_out, int out_size, void* d_ws, size_t ws_size,
                              hipStream_t stream) {
  const float* x = (const float*)d_in[0];  // (32, 4, 256, 256) f32
  const float* w = (const float*)d_in[1];  // (2, 4, 3) f32
  float* out = (float*)d_out;              // (32, 4, 128, 128) f32

  unsigned* keys = (unsigned*)d_ws;        // [0]=min key, [1]=max key
  float* M = (float*)((char*)d_ws + 256);  // 4 x 16 x 16 f32 observable matrices

  qc_init<<<1, 1, 0, stream>>>(keys);
  qc_minmax<<<512, 256, 0, stream>>>(x, keys, 524288);  // 2,097,152 floats as float4
  qc_build<<<1, 256, 0, stream>>>(w, M);
  qc_main<<<4096, 256, 0, stream>>>(x, M, keys, out);   // 4096 blocks * 8 waves * 16 samples
}